// DiagonalUpsample_70944269795383
// MI455X (gfx1250) — compile-verified
//
#include <hip/hip_runtime.h>

// DiagonalUpsample: checkerboard interleave of two (B,C,H,W) f32 tensors into
// (B,C,2H,2W).  Pure bandwidth problem: 302 MB total traffic -> ~13 us floor
// at 23.3 TB/s.  No FLOPs, so WMMA is not applicable; the optimal CDNA5 path
// is wide coalesced VMEM with non-temporal hints (streaming, output > L2).

#define UB 16
#define UC 3
#define UH 512
#define UW 512

typedef float v2f __attribute__((ext_vector_type(2)));
typedef float v4f __attribute__((ext_vector_type(4)));

// Each thread handles one float2 of each input row and emits one float4 to the
// even output row and one float4 to the odd output row.
//   input  float2 index t:  row = t / (W/2), col2 = t % (W/2)
//   flattened output rows: 2*row (even) and 2*row+1 (odd), width 2W
//   even row chunk: {d0, u0, d1, u1}   odd row chunk: {u0, d0, u1, d1}
__global__ __launch_bounds__(256) void DiagonalUpsample_kernel(
    const float* __restrict__ up,
    const float* __restrict__ down,
    float* __restrict__ out,
    int nvec2)  // total float2 units = B*C*H*(W/2)
{
    const int t = blockIdx.x * blockDim.x + threadIdx.x;
    if (t >= nvec2) return;

    const int row  = t >> 8;        // / (W/2) = /256
    const int col2 = t & 255;       // % 256

    // 64-bit loads: 32 lanes x 8B = 256B contiguous per wave per instruction.
    const v2f d = __builtin_nontemporal_load((const v2f*)down + t);
    const v2f u = __builtin_nontemporal_load((const v2f*)up + t);

    // Output addressing (floats). Row 2*row at width 2W=1024.
    // evenBase = (2*row)*1024 + 4*col2 ; oddBase = evenBase + 1024.
    const int evenBase = (row << 11) + (col2 << 2);

    const v4f e = { d.x, u.x, d.y, u.y };   // even row: d,u,d,u
    const v4f o = { u.x, d.x, u.y, d.y };   // odd  row: u,d,u,d

    // 128-bit stores: 32 lanes x 16B = 512B contiguous per wave per
    // instruction.  Non-temporal: output (201 MB) exceeds the 192 MB L2 and
    // is written exactly once.
    __builtin_nontemporal_store(e, (v4f*)(out + evenBase));
    __builtin_nontemporal_store(o, (v4f*)(out + evenBase + (UW * 2)));
}

extern "C" void kernel_launch(void* const* d_in, const int* in_sizes, int n_in,
                              void* d_out, int out_size, void* d_ws, size_t ws_size,
                              hipStream_t stream) {
    // setup_inputs() order: up_diagonal, down_diagonal  (both f32)
    const float* up   = (const float*)d_in[0];
    const float* down = (const float*)d_in[1];
    float*       out  = (float*)d_out;

    const int nvec2 = in_sizes[0] / 2;               // B*C*H*W/2 = 6,291,456
    const int block = 256;                           // 8 wave32 per block
    const int grid  = (nvec2 + block - 1) / block;   // 24576 blocks

    DiagonalUpsample_kernel<<<grid, block, 0, stream>>>(up, down, out, nvec2);
}